// LGCN_89240830477003
// MI455X (gfx1250) — compile-verified
//
#include <hip/hip_runtime.h>
#include <hip/hip_bf16.h>

#define NTOT   300000
#define RPv    16
#define NNODE  50000
#define EMBv   16
#define NCLS   10
#define EPSv   1e-6f

typedef __attribute__((ext_vector_type(2))) float v2f;
typedef __attribute__((ext_vector_type(8))) float v8f;

// ---------------------------------------------------------------------------
// K1: latent GEMM  z1 = nhots@Wl1+bl1, z2 = nhots@Wl2+bl2  (WMMA f32 16x16x4)
//     then per-row softmax -> latA (layer1), latB (layer2, relu(softmax))
//     one wave (32 threads) per 16-row tile of nhots.
//     K padded 50 -> 52; tail handled with clamped addresses + v_cndmask
//     (no EXEC-divergent predicated loads).
// ---------------------------------------------------------------------------
__global__ __launch_bounds__(32)
void k_latent(const float* __restrict__ nhots,
              const float* __restrict__ Wl1, const float* __restrict__ bl1,
              const float* __restrict__ Wl2, const float* __restrict__ bl2,
              float* __restrict__ latA, float* __restrict__ latB)
{
    __shared__ float zbuf[16][32];
    const int lane = threadIdx.x;
    const int t0   = blockIdx.x * 16;
    const int m    = lane & 15;          // M (row) index for A, N (col) for B
    const int hi   = lane >> 4;          // lane-half selects K pair

    const float* arow = &nhots[(size_t)(t0 + m) * 50];

    v8f c0 = {}; v8f c1 = {};
#pragma unroll
    for (int k0 = 0; k0 < 52; k0 += 4) {
        const int kb = k0 + (hi << 1);              // 0..50, even
        const int kc = (kb < 50) ? kb : 48;         // clamped, kc+1 <= 49
        const bool ok = (kb < 50);                  // only kb==50 is padding

        v2f a = *(const v2f*)(arow + kc);           // 8B-aligned (kc even)
        v2f b0, b1;
        b0.x = Wl1[kc * 16 + m];  b0.y = Wl1[(kc + 1) * 16 + m];
        b1.x = Wl2[kc * 16 + m];  b1.y = Wl2[(kc + 1) * 16 + m];
        if (!ok) { a.x = 0.f; a.y = 0.f; b0.x = 0.f; b0.y = 0.f; b1.x = 0.f; b1.y = 0.f; }

        c0 = __builtin_amdgcn_wmma_f32_16x16x4_f32(false, a, false, b0, (short)0, c0, false, false);
        c1 = __builtin_amdgcn_wmma_f32_16x16x4_f32(false, a, false, b1, (short)0, c1, false, false);
    }

    // C/D layout: element i -> row (i + 8*hi), col m
#pragma unroll
    for (int i = 0; i < 8; ++i) {
        const int row = i + (hi << 3);
        zbuf[row][m]      = c0[i] + bl1[m];
        zbuf[row][16 + m] = c1[i] + bl2[m];
    }
    __syncthreads();

    // softmax: lane -> (row = lane>>1, half = lane&1); 16 values each
    const int row = lane >> 1, half = lane & 1;
    const float* z = &zbuf[row][half * 16];
    float mx = z[0];
#pragma unroll
    for (int r = 1; r < 16; ++r) mx = fmaxf(mx, z[r]);
    float e[16], s = 0.f;
#pragma unroll
    for (int r = 0; r < 16; ++r) { e[r] = __expf(z[r] - mx); s += e[r]; }
    const float inv = 1.f / s;
    float* dst = half ? latB : latA;
#pragma unroll
    for (int r = 0; r < 16; ++r) {
        float v = e[r] * inv;
        if (half) v = fmaxf(v, 0.f);   // reference applies relu to lat2
        dst[(size_t)(t0 + row) * 16 + r] = v;
    }
}

// ---------------------------------------------------------------------------
// K2a: r==0 hot segments: colsum[0] += sum_t latA[t,0], rowsum[0] += sum_t latB[t,0]
// ---------------------------------------------------------------------------
__global__ void k_seg0(const float* __restrict__ latA, const float* __restrict__ latB,
                       float* __restrict__ colsum, float* __restrict__ rowsum)
{
    __shared__ float sA, sB;
    if (threadIdx.x == 0) { sA = 0.f; sB = 0.f; }
    __syncthreads();
    float a = 0.f, b = 0.f;
    for (int t = blockIdx.x * blockDim.x + threadIdx.x; t < NTOT;
         t += gridDim.x * blockDim.x) {
        a += latA[(size_t)t * 16];
        b += latB[(size_t)t * 16];
    }
    atomicAdd(&sA, a); atomicAdd(&sB, b);
    __syncthreads();
    if (threadIdx.x == 0) { atomicAdd(&colsum[0], sA); atomicAdd(&rowsum[0], sB); }
}

// ---------------------------------------------------------------------------
// K2b: segment sums for r >= 1 (scattered atomics, low contention)
// ---------------------------------------------------------------------------
__global__ void k_segsums(const float* __restrict__ latA, const float* __restrict__ latB,
                          const int* __restrict__ hcol, const int* __restrict__ vrow,
                          float* __restrict__ colsum, float* __restrict__ rowsum)
{
    const long i = (long)blockIdx.x * blockDim.x + threadIdx.x + NTOT;
    if (i >= (long)NTOT * RPv) return;
    const int r = (int)(i / NTOT);
    const int t = (int)(i - (long)r * NTOT);
    atomicAdd(&colsum[hcol[i]], latA[(size_t)t * 16 + r]);
    atomicAdd(&rowsum[vrow[i]], latB[(size_t)t * 16 + r]);
}

// ---------------------------------------------------------------------------
// K3: layer-1 SpMM: h[hrow] += (latA/colsum[hcol]) * weights1[hcol, :]
// ---------------------------------------------------------------------------
__global__ void k_spmm1(const float* __restrict__ latA,
                        const int* __restrict__ hrow, const int* __restrict__ hcol,
                        const float* __restrict__ colsum,
                        const float* __restrict__ w1, float* __restrict__ h)
{
    const long i = (long)blockIdx.x * blockDim.x + threadIdx.x;
    if (i >= (long)NTOT * RPv) return;
    const int r = (int)(i / NTOT);
    const int t = (int)(i - (long)r * NTOT);
    const int col = hcol[i];
    const float v = latA[(size_t)t * 16 + r] / fmaxf(colsum[col], EPSv);
    const float4* src = (const float4*)&w1[(size_t)col * 16];
    float* dst = &h[(size_t)hrow[i] * 16];
#pragma unroll
    for (int q = 0; q < 4; ++q) {
        const float4 x = src[q];
        atomicAdd(&dst[q * 4 + 0], v * x.x);
        atomicAdd(&dst[q * 4 + 1], v * x.y);
        atomicAdd(&dst[q * 4 + 2], v * x.z);
        atomicAdd(&dst[q * 4 + 3], v * x.w);
    }
}

// ---------------------------------------------------------------------------
// K4: h = relu(h + bias1)
// ---------------------------------------------------------------------------
__global__ void k_bias_relu(float* __restrict__ h, const float* __restrict__ bias1)
{
    const int i = blockIdx.x * blockDim.x + threadIdx.x;
    if (i >= NNODE * EMBv) return;
    h[i] = fmaxf(h[i] + bias1[i & 15], 0.f);
}

// ---------------------------------------------------------------------------
// K5a: r==0 hot row: h2[0, :] += sum_t (latB[t,0]/rowsum[0]) * h[vcol[t], :]
// ---------------------------------------------------------------------------
__global__ void k_spmm2_r0(const float* __restrict__ latB,
                           const float* __restrict__ rowsum,
                           const int* __restrict__ vcol,
                           const float* __restrict__ h, float* __restrict__ h2)
{
    __shared__ float acc[16];
    if (threadIdx.x < 16) acc[threadIdx.x] = 0.f;
    __syncthreads();
    const float invden = 1.f / fmaxf(rowsum[0], EPSv);
    float loc[16];
#pragma unroll
    for (int e = 0; e < 16; ++e) loc[e] = 0.f;
    for (int t = blockIdx.x * blockDim.x + threadIdx.x; t < NTOT;
         t += gridDim.x * blockDim.x) {
        const float v = latB[(size_t)t * 16] * invden;
        const float4* src = (const float4*)&h[(size_t)vcol[t] * 16];
#pragma unroll
        for (int q = 0; q < 4; ++q) {
            const float4 x = src[q];
            loc[q * 4 + 0] += v * x.x; loc[q * 4 + 1] += v * x.y;
            loc[q * 4 + 2] += v * x.z; loc[q * 4 + 3] += v * x.w;
        }
    }
#pragma unroll
    for (int e = 0; e < 16; ++e) atomicAdd(&acc[e], loc[e]);
    __syncthreads();
    if (threadIdx.x < 16) atomicAdd(&h2[threadIdx.x], acc[threadIdx.x]);
}

// ---------------------------------------------------------------------------
// K5b: layer-2 SpMM, r >= 1: h2[vrow] += (latB/rowsum[vrow]) * h[vcol, :]
// ---------------------------------------------------------------------------
__global__ void k_spmm2(const float* __restrict__ latB,
                        const int* __restrict__ vrow, const int* __restrict__ vcol,
                        const float* __restrict__ rowsum,
                        const float* __restrict__ h, float* __restrict__ h2)
{
    const long i = (long)blockIdx.x * blockDim.x + threadIdx.x + NTOT;
    if (i >= (long)NTOT * RPv) return;
    const int r = (int)(i / NTOT);
    const int t = (int)(i - (long)r * NTOT);
    const int rw = vrow[i];
    const float v = latB[(size_t)t * 16 + r] / fmaxf(rowsum[rw], EPSv);
    const float4* src = (const float4*)&h[(size_t)vcol[i] * 16];
    float* dst = &h2[(size_t)rw * 16];
#pragma unroll
    for (int q = 0; q < 4; ++q) {
        const float4 x = src[q];
        atomicAdd(&dst[q * 4 + 0], v * x.x);
        atomicAdd(&dst[q * 4 + 1], v * x.y);
        atomicAdd(&dst[q * 4 + 2], v * x.z);
        atomicAdd(&dst[q * 4 + 3], v * x.w);
    }
}

// ---------------------------------------------------------------------------
// K6: logits[n,c] = sum_r h2[r, n, :] @ W2[r]  + bias2   (WMMA f32 16x16x4)
//     one wave per 16-node tile. h2 tiles (1 KB, contiguous) are double-
//     buffered in LDS via global_load_async_to_lds_b128 (async DMA path),
//     A-fragments read back with ds_load_b64. 16 r-iters x 4 k-steps.
// ---------------------------------------------------------------------------
__global__ __launch_bounds__(32)
void k_einsum(const float* __restrict__ h2, const float* __restrict__ W2,
              const float* __restrict__ bias2, float* __restrict__ out)
{
    __shared__ __align__(16) float abuf[2][256];   // 2 x 1KB tile buffers
    const int lane = threadIdx.x;
    const int n0   = blockIdx.x * 16;
    const int m    = lane & 15;        // node row for A, class col for B
    const int hi   = lane >> 4;
    const int mc   = (m < NCLS) ? m : (NCLS - 1);  // clamped class column

    // defined-init LDS (avoid reads-of-never-written UB), then drain DS ops
    abuf[0][lane * 8 + 0] = 0.f;  // compiler vectorizes; 8 floats per lane x2 bufs
#pragma unroll
    for (int q = 1; q < 8; ++q) abuf[0][lane * 8 + q] = 0.f;
#pragma unroll
    for (int q = 0; q < 8; ++q) abuf[1][lane * 8 + q] = 0.f;
    asm volatile("s_wait_dscnt 0x0" ::: "memory");

    const unsigned ldsbase = (unsigned)(size_t)&abuf[0][0];

    // async-copy one 1KB tile (tile r, buffer b): 2 x b128 per lane
#define ISSUE_TILE(rr, bb)                                                      \
    do {                                                                        \
        const char* gp = (const char*)(h2 + ((size_t)(rr) * NNODE + n0) * 16)   \
                         + (size_t)lane * 16;                                   \
        unsigned lp = ldsbase + (unsigned)((bb) * 1024) + (unsigned)lane * 16;  \
        asm volatile("global_load_async_to_lds_b128 %0, %1, off"                \
                     :: "v"(lp), "v"(gp) : "memory");                           \
        asm volatile("global_load_async_to_lds_b128 %0, %1, off"                \
                     :: "v"(lp + 512u), "v"(gp + 512) : "memory");              \
    } while (0)

    ISSUE_TILE(0, 0);

    v8f acc = {};
    for (int r = 0; r < RPv; ++r) {
        if (r + 1 < RPv) {
            ISSUE_TILE(r + 1, (r + 1) & 1);
            asm volatile("s_wait_asynccnt 0x2" ::: "memory");  // tile r complete
        } else {
            asm volatile("s_wait_asynccnt 0x0" ::: "memory");
        }
        const float* ab     = &abuf[r & 1][0];
        const float* b_base = &W2[(size_t)r * EMBv * NCLS];
#pragma unroll
        for (int ks = 0; ks < 4; ++ks) {
            const int kb = ks * 4 + (hi << 1);
            v2f a = *(const v2f*)&ab[m * 16 + kb];             // ds_load_b64
            v2f b;
            b.x = b_base[kb * NCLS + mc];
            b.y = b_base[(kb + 1) * NCLS + mc];
            if (m >= NCLS) { b.x = 0.f; b.y = 0.f; }           // v_cndmask, no exec div
            acc = __builtin_amdgcn_wmma_f32_16x16x4_f32(false, a, false, b, (short)0, acc, false, false);
        }
    }
#undef ISSUE_TILE

    if (m < NCLS) {
#pragma unroll
        for (int i = 0; i < 8; ++i) {
            const int row = i + (hi << 3);
            out[(size_t)(n0 + row) * NCLS + m] = acc[i] + bias2[m];
        }
    }
}

// ---------------------------------------------------------------------------
extern "C" void kernel_launch(void* const* d_in, const int* in_sizes, int n_in,
                              void* d_out, int out_size, void* d_ws, size_t ws_size,
                              hipStream_t stream)
{
    const float* nhots = (const float*)d_in[0];
    const int*   hrow  = (const int*)  d_in[1];
    const int*   hcol  = (const int*)  d_in[2];
    const int*   vrow  = (const int*)  d_in[3];
    const int*   vcol  = (const int*)  d_in[4];
    const float* Wl1   = (const float*)d_in[5];
    const float* bl1   = (const float*)d_in[6];
    const float* Wl2   = (const float*)d_in[7];
    const float* bl2   = (const float*)d_in[8];
    const float* w1    = (const float*)d_in[9];   // (RP, N, EMB) flat = (RP*N, EMB)
    const float* bias1 = (const float*)d_in[10];
    const float* W2    = (const float*)d_in[11];  // (RP, EMB, NCLS)
    const float* bias2 = (const float*)d_in[12];
    float* out = (float*)d_out;

    // workspace layout
    char* ws = (char*)d_ws;
    const size_t szLat = (size_t)NTOT * RPv * sizeof(float);         // 19.2 MB
    const size_t szSeg = (size_t)RPv * NNODE * sizeof(float);        // 3.2 MB
    const size_t szH   = (size_t)NNODE * EMBv * sizeof(float);       // 3.2 MB
    const size_t szH2  = (size_t)RPv * NNODE * EMBv * sizeof(float); // 51.2 MB
    float* latA   = (float*)(ws);                   ws += szLat;
    float* latB   = (float*)(ws);                   ws += szLat;
    float* colsum = (float*)(ws);                   ws += szSeg;
    float* rowsum = (float*)(ws);                   ws += szSeg;
    float* h      = (float*)(ws);                   ws += szH;
    float* h2     = (float*)(ws);                   ws += szH2;

    hipMemsetAsync(colsum, 0, szSeg, stream);
    hipMemsetAsync(rowsum, 0, szSeg, stream);
    hipMemsetAsync(h,      0, szH,   stream);
    hipMemsetAsync(h2,     0, szH2,  stream);

    // K1: latent GEMM + softmax
    k_latent<<<NTOT / 16, 32, 0, stream>>>(nhots, Wl1, bl1, Wl2, bl2, latA, latB);

    // K2: segment sums
    k_seg0<<<512, 256, 0, stream>>>(latA, latB, colsum, rowsum);
    {
        const long n = (long)NTOT * (RPv - 1);
        k_segsums<<<(unsigned)((n + 255) / 256), 256, 0, stream>>>(latA, latB, hcol, vrow,
                                                                   colsum, rowsum);
    }

    // K3: layer-1 normalize + scatter SpMM
    {
        const long n = (long)NTOT * RPv;
        k_spmm1<<<(unsigned)((n + 255) / 256), 256, 0, stream>>>(latA, hrow, hcol, colsum,
                                                                 w1, h);
    }

    // K4: bias + relu
    k_bias_relu<<<(NNODE * EMBv + 255) / 256, 256, 0, stream>>>(h, bias1);

    // K5: layer-2 normalize + scatter SpMM
    k_spmm2_r0<<<512, 256, 0, stream>>>(latB, rowsum, vcol, h, h2);
    {
        const long n = (long)NTOT * (RPv - 1);
        k_spmm2<<<(unsigned)((n + 255) / 256), 256, 0, stream>>>(latB, vrow, vcol, rowsum,
                                                                 h, h2);
    }

    // K6: final einsum via WMMA (async-LDS double-buffered A tiles)
    k_einsum<<<NNODE / 16, 32, 0, stream>>>(h2, W2, bias2, out);
}